// span_pooler_40097814675703
// MI455X (gfx1250) — compile-verified
//
#include <hip/hip_runtime.h>

// ---------------------------------------------------------------------------
// Problem constants (from reference): B=8,S=4,L=512,H=768; N_SPANS=4096,
// T_MAX=16, N_PAIRS=16384. reprs row = [first(768) | mean(768) | last(768)].
// ---------------------------------------------------------------------------
#define NSPANS 4096
#define TMAX   16
#define HDIM   768
#define SL     2048          // S*L
#define REPR   2304          // 3*H
#define NPAIRS 16384

typedef __attribute__((ext_vector_type(2))) float v2f;
typedef __attribute__((ext_vector_type(8))) float v8f;

// ---------------------------------------------------------------------------
// Kernel 1: one block (256 threads = 8 waves) per span.
//  - first/last: straight float4 copies (b128 loads/stores).
//  - mean: V_WMMA_F32_16X16X4_F32, weights folded into A, B = all-ones so the
//    result is independent of B's VGPR layout. 4 chained WMMAs give K=16.
// ---------------------------------------------------------------------------
__global__ __launch_bounds__(256) void span_reprs_kernel(
    const float* __restrict__ hidden,
    const int*   __restrict__ span_doc,
    const int*   __restrict__ span_tok,
    const int*   __restrict__ span_len,
    float*       __restrict__ reprs)
{
    const int s   = blockIdx.x;
    const int tid = threadIdx.x;

    __shared__ size_t base[TMAX];   // float offset of each gathered token row
    __shared__ float  wgt[TMAX];    // mask/len weights

    const int len = span_len[s];
    if (tid < TMAX) {
        const int doc = span_doc[s];
        const int tok = span_tok[s * TMAX + tid];
        base[tid] = ((size_t)doc * SL + (size_t)tok) * HDIM;
        wgt[tid]  = (tid < len) ? (1.0f / (float)len) : 0.0f;
    }
    __syncthreads();

    float* row = reprs + (size_t)s * REPR;

    // ---- first / last (b128 path) -----------------------------------------
    {
        const float4* fsrc = (const float4*)(hidden + base[0]);
        const float4* lsrc = (const float4*)(hidden + base[len - 1]);
        float4* fdst = (float4*)row;                 // first -> [0, 768)
        float4* ldst = (float4*)(row + 2 * HDIM);    // last  -> [1536, 2304)
        for (int i = tid; i < HDIM / 4; i += 256) {
            fdst[i] = fsrc[i];
            ldst[i] = lsrc[i];
        }
    }

    // ---- masked mean via chained V_WMMA_F32_16X16X4_F32 -------------------
    // Per 16-wide h-tile: D(16x16) = sum_c A_c(16x4) x Ones(4x16)
    //   A_c[m,k] = wgt[4c+k] * g[tok_{4c+k}][h0+m]   (weights folded into A)
    // => every column of D equals mean[h0 .. h0+15].
    //
    // Documented 32-bit A 16x4 layout:
    //   lanes 0-15 : M=lane,  VGPR0=K0, VGPR1=K1
    //   lanes 16-31: M=lane-16, VGPR0=K2, VGPR1=K3
    const int lane  = tid & 31;
    const int wave  = tid >> 5;
    const int m     = lane & 15;
    const int khalf = lane >> 4;     // 0: K0/K1, 1: K2/K3

    const v2f ones = {1.0f, 1.0f};

    for (int tile = wave; tile < HDIM / 16; tile += 8) {
        const int h = tile * 16 + m;
        v8f acc = {0.f, 0.f, 0.f, 0.f, 0.f, 0.f, 0.f, 0.f};
#pragma unroll
        for (int c = 0; c < 4; ++c) {
            const int k0 = 4 * c + 2 * khalf;
            v2f a;
            a.x = wgt[k0]     * hidden[base[k0]     + h];
            a.y = wgt[k0 + 1] * hidden[base[k0 + 1] + h];
            // 8 args: (neg_a, A, neg_b, B, c_mod, C, reuse_a, reuse_b)
            acc = __builtin_amdgcn_wmma_f32_16x16x4_f32(
                false, a, false, ones, (short)0, acc, false, false);
        }
        // Documented C/D layout: VGPR r holds M=r (lanes 0-15) / M=r+8
        // (lanes 16-31). All 16 columns identical, so lanes 0-7 emit
        // h0+0..7 and lanes 16-23 emit h0+8..15 (one value each).
        const int e = lane & 7;
        float val = acc[0];
        val = (e == 1) ? acc[1] : val;
        val = (e == 2) ? acc[2] : val;
        val = (e == 3) ? acc[3] : val;
        val = (e == 4) ? acc[4] : val;
        val = (e == 5) ? acc[5] : val;
        val = (e == 6) ? acc[6] : val;
        val = (e == 7) ? acc[7] : val;
        if ((lane & 15) < 8) {
            row[HDIM + tile * 16 + khalf * 8 + e] = val;   // mean -> [768,1536)
        }
    }
}

// ---------------------------------------------------------------------------
// Kernel 2: one block per pair. Pure b128 copy-expand; reprs (37.7 MB) is
// L2-resident so the 288 MB of reads stay on-chip; the 288 MB output write
// is the HBM roofline (~12.4 us at 23.3 TB/s).
// ---------------------------------------------------------------------------
__global__ __launch_bounds__(256) void pair_expand_kernel(
    const float* __restrict__ reprs,
    const int*   __restrict__ pair_i,
    const int*   __restrict__ pair_j,
    float*       __restrict__ out)
{
    const int p   = blockIdx.x;
    const int tid = threadIdx.x;
    const int pi  = pair_i[p];
    const int pj  = pair_j[p];

    const float4* si  = (const float4*)(reprs + (size_t)pi * REPR);
    const float4* sj  = (const float4*)(reprs + (size_t)pj * REPR);
    float4*       dst = (float4*)(out + (size_t)p * (2 * REPR));

    const int n4 = REPR / 4;   // 576 float4 per repr row
    for (int i = tid; i < n4; i += 256) {
        dst[i]      = si[i];
        dst[n4 + i] = sj[i];
    }
}

// ---------------------------------------------------------------------------
// Launcher: reprs staged in d_ws (4096*2304*4 = 37,748,736 bytes), then
// pair expansion. Both on `stream`; stream order provides the dependency.
// ---------------------------------------------------------------------------
extern "C" void kernel_launch(void* const* d_in, const int* in_sizes, int n_in,
                              void* d_out, int out_size, void* d_ws, size_t ws_size,
                              hipStream_t stream) {
    const float* hidden   = (const float*)d_in[0];
    const int*   span_doc = (const int*)d_in[1];
    const int*   span_tok = (const int*)d_in[2];
    const int*   span_len = (const int*)d_in[3];
    const int*   pair_i   = (const int*)d_in[4];
    const int*   pair_j   = (const int*)d_in[5];
    float*       out      = (float*)d_out;
    float*       reprs    = (float*)d_ws;

    span_reprs_kernel<<<NSPANS, 256, 0, stream>>>(hidden, span_doc, span_tok,
                                                  span_len, reprs);
    pair_expand_kernel<<<NPAIRS, 256, 0, stream>>>(reprs, pair_i, pair_j, out);
}